// GatedMultiHeadGATLayer_16363825398384
// MI455X (gfx1250) — compile-verified
//
#include <hip/hip_runtime.h>
#include <cstdint>

#define NN 40000
#define EE 640000
#define DD 128
#define HH 4
#define NEG 0.01f

typedef float v2f __attribute__((ext_vector_type(2)));
typedef float v8f __attribute__((ext_vector_type(8)));

__device__ __forceinline__ v8f wmma_f32(v2f a, v2f b, v8f c) {
  // V_WMMA_F32_16X16X4_F32: D = A(16x4) x B(4x16) + C(16x16), full fp32
  return __builtin_amdgcn_wmma_f32_16x16x4_f32(false, a, false, b, (short)0, c, false, false);
}

__device__ __forceinline__ float lrelu(float v) { return v > 0.f ? v : NEG * v; }
__device__ __forceinline__ float sigf(float x) { return 1.f / (1.f + __expf(-x)); }

// order-preserving float<->uint transform for atomic max on floats of any sign
__device__ __forceinline__ unsigned ordf(float f) {
  unsigned u = __float_as_uint(f);
  return (u & 0x80000000u) ? ~u : (u | 0x80000000u);
}
__device__ __forceinline__ float unordf(unsigned v) {
  return (v & 0x80000000u) ? __uint_as_float(v ^ 0x80000000u) : __uint_as_float(~v);
}

__global__ void fill_u32(unsigned* __restrict__ p, unsigned v, int n) {
  int i = blockIdx.x * 256 + threadIdx.x;
  if (i < n) p[i] = v;
}

// C[head][r][0..127] = lrelu( sum_k A[r][k] * W[head][c][k] )
// One wave per 16-row strip; 8 accumulator tiles cover 128 output cols.
// Weights staged through LDS in 64-wide K chunks, pitch 66 (bank-conflict free b64 reads).
__global__ __launch_bounds__(128) void gemm_lrelu_kernel(
    const float* __restrict__ A, int lda,
    const float* __restrict__ W, int K, long wHeadStride,
    float* __restrict__ C, long cHeadStride) {
  __shared__ __align__(16) float lw[128 * 66];
  const float* Wh = W + (long)blockIdx.y * wHeadStride;
  float* Ch = C + (long)blockIdx.y * cHeadStride;
  int tid = threadIdx.x;
  int wave = tid >> 5, lane = tid & 31;
  int r0 = (blockIdx.x * 4 + wave) * 16;
  int half = lane >> 4, n = lane & 15;

  v8f acc[8];
#pragma unroll
  for (int t = 0; t < 8; ++t) acc[t] = (v8f){0.f, 0.f, 0.f, 0.f, 0.f, 0.f, 0.f, 0.f};

  for (int kc = 0; kc < K; kc += 64) {
    __syncthreads();
    for (int idx = tid; idx < 128 * 64; idx += 128) {
      int o = idx >> 6, kk = idx & 63;
      lw[o * 66 + kk] = Wh[(long)o * K + kc + kk];
    }
    __syncthreads();
#pragma unroll 4
    for (int k = 0; k < 64; k += 4) {
      int kb = k + 2 * half;
      v2f a = *(const v2f*)&A[(long)(r0 + n) * lda + kc + kb];
#pragma unroll
      for (int t = 0; t < 8; ++t) {
        v2f b = *(const v2f*)&lw[(t * 16 + n) * 66 + kb];
        acc[t] = wmma_f32(a, b, acc[t]);
      }
    }
  }
#pragma unroll
  for (int t = 0; t < 8; ++t)
#pragma unroll
    for (int j = 0; j < 8; ++j)
      Ch[(long)(r0 + j + 8 * half) * DD + t * 16 + n] = lrelu(acc[t][j]);
}

// a_src[hd][node] = z[hd][node] . attnW[hd][0:128]; a_dst uses attnW[hd][128:256]
__global__ __launch_bounds__(256) void attn_kernel(
    const float* __restrict__ Z, long zHeadStride,
    const float* __restrict__ attnW, long aHeadStride,
    float* __restrict__ asrc, float* __restrict__ adst, int Nn) {
  int hd = blockIdx.y;
  int wave = threadIdx.x >> 5, lane = threadIdx.x & 31;
  long node = (long)blockIdx.x * 8 + wave;
  const float* zp = Z + (long)hd * zHeadStride + node * DD;
  const float* aw = attnW + (long)hd * aHeadStride;
  float s1 = 0.f, s2 = 0.f;
#pragma unroll
  for (int q = 0; q < 4; ++q) {
    int c = lane + q * 32;
    float zv = zp[c];
    s1 += zv * aw[c];
    s2 += zv * aw[128 + c];
  }
#pragma unroll
  for (int off = 16; off; off >>= 1) {
    s1 += __shfl_xor(s1, off, 32);
    s2 += __shfl_xor(s2, off, 32);
  }
  if (lane == 0) {
    asrc[(long)hd * Nn + node] = s1;
    adst[(long)hd * Nn + node] = s2;
  }
}

// per-edge score ne = m_w * lrelu((a_src+a_dst) * pd * edge_w); segment max via ordered-uint atomicMax
__global__ void edge_pass1(const float* __restrict__ asrc, const float* __restrict__ adst,
                           const float* __restrict__ pd, const int* __restrict__ src,
                           const int* __restrict__ dst, const float* __restrict__ edge_w,
                           const float* __restrict__ m_w, float* __restrict__ neb,
                           unsigned* __restrict__ nmax, int Nn, int Ee) {
  int hd = blockIdx.y;
  int e = blockIdx.x * 256 + threadIdx.x;
  int s = src[e], d = dst[e];
  float a = asrc[(long)hd * Nn + s] + adst[(long)hd * Nn + d];
  float t = a * (pd[e] * edge_w[hd]);
  t = t > 0.f ? t : NEG * t;
  float ne = t * m_w[hd];
  neb[(long)hd * Ee + e] = ne;
  atomicMax(&nmax[(long)hd * Nn + d], ordf(ne));
}

// ex = exp(ne - max[dst]); segment sum via atomicAdd
__global__ void edge_pass2(float* __restrict__ neb, const int* __restrict__ dst,
                           const unsigned* __restrict__ nmax, float* __restrict__ nsum,
                           int Nn, int Ee) {
  int hd = blockIdx.y;
  int e = blockIdx.x * 256 + threadIdx.x;
  int d = dst[e];
  float ne = neb[(long)hd * Ee + e];
  float mx = unordf(nmax[(long)hd * Nn + d]);
  float ex = __expf(ne - mx);
  neb[(long)hd * Ee + e] = ex;
  atomicAdd(&nsum[(long)hd * Nn + d], ex);
}

// agg[dst][hd*colOff + c] += ex * z[hd][src][c]  (one wave per (edge,head), float4 gather)
__global__ __launch_bounds__(256) void edge_agg(
    const float* __restrict__ Z, long zHeadStride, const float* __restrict__ exb,
    const int* __restrict__ src, const int* __restrict__ dst,
    float* __restrict__ agg, int aggPitch, int colOffPerHead, int Nn, int Ee) {
  int hd = blockIdx.y;
  int wave = threadIdx.x >> 5, lane = threadIdx.x & 31;
  long e = (long)blockIdx.x * 8 + wave;
  float ex = exb[(long)hd * Ee + e];
  int s = src[e], d = dst[e];
  float4 z4 = *(const float4*)&Z[(long)hd * zHeadStride + (long)s * DD + lane * 4];
  float* o = &agg[(long)d * aggPitch + hd * colOffPerHead + lane * 4];
  atomicAdd(o + 0, ex * z4.x);
  atomicAdd(o + 1, ex * z4.y);
  atomicAdd(o + 2, ex * z4.z);
  atomicAdd(o + 3, ex * z4.w);
}

// x = lrelu(agg / max(sum,>0?sum:1)); pitchShift: 9 (512 cols) or 7 (128 cols)
__global__ void normalize_lrelu(float* __restrict__ agg, const float* __restrict__ nsum,
                                int pitchShift, int Nn) {
  long idx = (long)blockIdx.x * 256 + threadIdx.x;
  long row = idx >> pitchShift;
  int col = (int)(idx & ((1 << pitchShift) - 1));
  int hd = col >> 7;
  float s = nsum[(long)hd * Nn + row];
  float denom = s > 0.f ? s : 1.f;
  agg[idx] = lrelu(agg[idx] / denom);
}

// GRUCell + final lrelu. One wave per 16x16 output tile; six WMMA gate accumulators.
__global__ __launch_bounds__(128) void gru_kernel(
    const float* __restrict__ X, const float* __restrict__ Hp,
    const float* __restrict__ Wih, const float* __restrict__ Whh,
    const float* __restrict__ bih, const float* __restrict__ bhh,
    float* __restrict__ out) {
  int tid = threadIdx.x, wave = tid >> 5, lane = tid & 31;
  int w = blockIdx.x * 4 + wave;
  int r0 = (w >> 3) * 16, c0 = (w & 7) * 16;
  int half = lane >> 4, n = lane & 15;
  v8f ir = (v8f){0.f,0.f,0.f,0.f,0.f,0.f,0.f,0.f};
  v8f iz = ir, in_ = ir, hr = ir, hz = ir, hn = ir;
#pragma unroll 4
  for (int k = 0; k < 128; k += 4) {
    int kb = k + 2 * half;
    v2f ax = *(const v2f*)&X[(long)(r0 + n) * DD + kb];
    v2f ah = *(const v2f*)&Hp[(long)(r0 + n) * DD + kb];
    v2f b;
    b = *(const v2f*)&Wih[(long)(c0 + n) * DD + kb];        ir  = wmma_f32(ax, b, ir);
    b = *(const v2f*)&Wih[(long)(128 + c0 + n) * DD + kb];  iz  = wmma_f32(ax, b, iz);
    b = *(const v2f*)&Wih[(long)(256 + c0 + n) * DD + kb];  in_ = wmma_f32(ax, b, in_);
    b = *(const v2f*)&Whh[(long)(c0 + n) * DD + kb];        hr  = wmma_f32(ah, b, hr);
    b = *(const v2f*)&Whh[(long)(128 + c0 + n) * DD + kb];  hz  = wmma_f32(ah, b, hz);
    b = *(const v2f*)&Whh[(long)(256 + c0 + n) * DD + kb];  hn  = wmma_f32(ah, b, hn);
  }
  int c = c0 + n;
  float b_ir = bih[c], b_iz = bih[128 + c], b_in = bih[256 + c];
  float b_hr = bhh[c], b_hz = bhh[128 + c], b_hn = bhh[256 + c];
#pragma unroll
  for (int j = 0; j < 8; ++j) {
    int row = r0 + j + 8 * half;
    float rg = sigf(ir[j] + b_ir + hr[j] + b_hr);
    float zg = sigf(iz[j] + b_iz + hz[j] + b_hz);
    float ng = tanhf(in_[j] + b_in + rg * (hn[j] + b_hn));
    float hp = Hp[(long)row * DD + c];
    float o = (1.f - zg) * ng + zg * hp;
    out[(long)row * DD + c] = lrelu(o);
  }
}

extern "C" void kernel_launch(void* const* d_in, const int* in_sizes, int n_in,
                              void* d_out, int out_size, void* d_ws, size_t ws_size,
                              hipStream_t stream) {
  (void)in_sizes; (void)n_in; (void)out_size; (void)ws_size;
  const float* h     = (const float*)d_in[0];
  const float* pd    = (const float*)d_in[1];
  const int*   src   = (const int*)d_in[2];
  const int*   dst   = (const int*)d_in[3];
  const float* fcW   = (const float*)d_in[4];
  const float* attnW = (const float*)d_in[5];
  const float* edgeW = (const float*)d_in[6];
  const float* mW    = (const float*)d_in[7];
  const float* ofcW  = (const float*)d_in[8];
  const float* oattnW= (const float*)d_in[9];
  const float* oedgeW= (const float*)d_in[10];
  const float* omW   = (const float*)d_in[11];
  const float* Wih   = (const float*)d_in[12];
  const float* Whh   = (const float*)d_in[13];
  const float* bih   = (const float*)d_in[14];
  const float* bhh   = (const float*)d_in[15];
  float* out = (float*)d_out;

  float* ws = (float*)d_ws;
  const long ND = (long)NN * DD;            // 5.12M floats
  float*    z    = ws;                      // 4*ND : per-head z
  float*    agg  = ws + 4 * ND;             // 4*ND : x = [N][512]
  float*    asrc = ws + 8 * ND;             // 4*N
  float*    adst = asrc + 4 * NN;           // 4*N
  float*    nsum = adst + 4 * NN;           // 4*N
  unsigned* nmax = (unsigned*)(nsum + 4 * NN); // 4*N
  float*    exb  = (float*)(nmax + 4 * NN); // 4*E
  // stage-2 scratch aliases the dead head-z region (safe: stream-ordered)
  float*    z2    = z;                      // ND
  float*    agg2  = z + ND;                 // ND
  float*    a2s   = z + 2 * ND;             // N
  float*    a2d   = a2s + NN;               // N
  float*    nsum2 = a2d + NN;               // N
  unsigned* nmax2 = (unsigned*)(nsum2 + NN);// N
  float*    exb2  = (float*)(nmax2 + NN);   // E

  const unsigned ORDNI = 0x007FFFFFu;       // ordf(-inf)

  // ---- stage 1: per-head GAT ----
  fill_u32<<<(4 * (int)ND + 255) / 256, 256, 0, stream>>>((unsigned*)agg, 0u, 4 * (int)ND);
  fill_u32<<<(4 * NN + 255) / 256, 256, 0, stream>>>((unsigned*)nsum, 0u, 4 * NN);
  fill_u32<<<(4 * NN + 255) / 256, 256, 0, stream>>>(nmax, ORDNI, 4 * NN);

  gemm_lrelu_kernel<<<dim3(NN / 64, HH), 128, 0, stream>>>(h, DD, fcW, DD, (long)DD * DD, z, ND);
  attn_kernel<<<dim3(NN / 8, HH), 256, 0, stream>>>(z, ND, attnW, 2L * DD, asrc, adst, NN);
  edge_pass1<<<dim3(EE / 256, HH), 256, 0, stream>>>(asrc, adst, pd, src, dst, edgeW, mW, exb, nmax, NN, EE);
  edge_pass2<<<dim3(EE / 256, HH), 256, 0, stream>>>(exb, dst, nmax, nsum, NN, EE);
  edge_agg<<<dim3(EE / 8, HH), 256, 0, stream>>>(z, ND, exb, src, dst, agg, HH * DD, DD, NN, EE);
  normalize_lrelu<<<(NN * HH * DD) / 256, 256, 0, stream>>>(agg, nsum, 9, NN);

  // ---- stage 2: output GAT (heads=1) ----
  fill_u32<<<((int)ND + 255) / 256, 256, 0, stream>>>((unsigned*)agg2, 0u, (int)ND);
  fill_u32<<<(NN + 255) / 256, 256, 0, stream>>>((unsigned*)nsum2, 0u, NN);
  fill_u32<<<(NN + 255) / 256, 256, 0, stream>>>(nmax2, ORDNI, NN);

  gemm_lrelu_kernel<<<dim3(NN / 64, 1), 128, 0, stream>>>(agg, HH * DD, ofcW, HH * DD, 0L, z2, 0L);
  attn_kernel<<<dim3(NN / 8, 1), 256, 0, stream>>>(z2, 0L, oattnW, 0L, a2s, a2d, NN);
  edge_pass1<<<dim3(EE / 256, 1), 256, 0, stream>>>(a2s, a2d, pd, src, dst, oedgeW, omW, exb2, nmax2, NN, EE);
  edge_pass2<<<dim3(EE / 256, 1), 256, 0, stream>>>(exb2, dst, nmax2, nsum2, NN, EE);
  edge_agg<<<dim3(EE / 8, 1), 256, 0, stream>>>(z2, 0L, exb2, src, dst, agg2, DD, 0, NN, EE);
  normalize_lrelu<<<(NN * DD) / 256, 256, 0, stream>>>(agg2, nsum2, 7, NN);

  // ---- stage 3: GRU + final lrelu ----
  gru_kernel<<<(NN / 16) * 8 / 4, 128, 0, stream>>>(agg2, h, Wih, Whh, bih, bhh, out);
}